// FrameContinuityLoss_21320217657924
// MI455X (gfx1250) — compile-verified
//
#include <hip/hip_runtime.h>
#include <hip/hip_bf16.h>
#include <stdint.h>

#define NC 32
#define NB 128
#define NW 8192

// ---------------- Kernel 1: argmax over 32 classes ----------------
// 128*8192 = 1,048,576 positions. Tile = 256 positions = 32KB of f32.
// Double-buffered async global->LDS staging; scattered during the async
// copy into [quad][pos] layout so ds_load_b128 reads are conflict-free.
#define TILE_POS 256
#define K1_THREADS 256
#define TILES_PER_BLOCK 4
#define K1_BLOCKS ((NB * NW) / (TILE_POS * TILES_PER_BLOCK)) // 1024

__global__ __launch_bounds__(K1_THREADS)
void fcl_argmax_kernel(const float* __restrict__ pred,
                       unsigned char* __restrict__ cls)
{
    __shared__ float4 buf[2][8][TILE_POS];   // 64 KB, [stage][class-quad][pos]
    const int tid = threadIdx.x;
    const long long pos0 = (long long)blockIdx.x * (TILE_POS * TILES_PER_BLOCK);

    auto issue = [&](int stage, int t) {
        // tile base in global memory (uniform per block -> SGPR base)
        const char* gbase =
            (const char*)(pred + (pos0 + (long long)t * TILE_POS) * NC);
        #pragma unroll
        for (int k = 0; k < 8; ++k) {
            int g = tid + k * K1_THREADS;            // chunk id 0..2047 (16B each)
            int p = g >> 3;                          // position within tile
            int q = g & 7;                           // class quad 0..7
            unsigned lds_off = (unsigned)(uintptr_t)&buf[stage][q][p];
            unsigned voff    = (unsigned)g * 16u;    // byte offset from gbase
            asm volatile("global_load_async_to_lds_b128 %0, %1, %2"
                         :
                         : "v"(lds_off), "v"(voff), "s"(gbase)
                         : "memory");
        }
    };

    issue(0, 0);
    issue(1, 1);

    for (int t = 0; t < TILES_PER_BLOCK; ++t) {
        const int stage = t & 1;
        if (t < TILES_PER_BLOCK - 1) {
            // one more tile (8 async loads) still in flight is OK
            asm volatile("s_wait_asynccnt 0x8" ::: "memory");
        } else {
            asm volatile("s_wait_asynccnt 0x0" ::: "memory");
        }
        __syncthreads();   // every wave has drained its tile-t loads

        float best; int bi;
        {
            float4 v = buf[stage][0][tid];
            best = v.x; bi = 0;
            if (v.y > best) { best = v.y; bi = 1; }
            if (v.z > best) { best = v.z; bi = 2; }
            if (v.w > best) { best = v.w; bi = 3; }
        }
        #pragma unroll
        for (int q = 1; q < 8; ++q) {
            float4 v = buf[stage][q][tid];
            const int c0 = q * 4;
            if (v.x > best) { best = v.x; bi = c0 + 0; }
            if (v.y > best) { best = v.y; bi = c0 + 1; }
            if (v.z > best) { best = v.z; bi = c0 + 2; }
            if (v.w > best) { best = v.w; bi = c0 + 3; }
        }
        cls[pos0 + (long long)t * TILE_POS + tid] = (unsigned char)bi;

        __syncthreads();   // all reads of `stage` done before async reuse
        if (t + 2 < TILES_PER_BLOCK) issue(stage, t + 2);
    }
}

// ---------------- Kernel 2: per-(row,class) max run length ----------
// One block per batch row; 32 waves, each handling a 256-position chunk.
// lane == class (wave32 == 32 classes). Associative combine of
// (prefix, suffix, best, full) tuples folds the 32 chunks.
// Writes a deterministic per-row sum of squared diffs (no atomics).
#define K2_THREADS 1024
#define CHUNK (NW / (K2_THREADS / 32))   // 256

__global__ __launch_bounds__(K2_THREADS)
void fcl_runlen_kernel(const unsigned char* __restrict__ cls,
                       const long long* __restrict__ tgt,
                       float* __restrict__ rowsum)
{
    __shared__ int tup[2][32][NC][4];    // [stream][chunk][class][pref,suff,best,full]
    const int lane = threadIdx.x & 31;
    const int wave = threadIdx.x >> 5;
    const int row  = blockIdx.x;
    const long long base = (long long)row * NW + (long long)wave * CHUNK;

    int prun = 0, pbest = 0, ppref = 0, pfull = 1;
    int trun = 0, tbest = 0, tpref = 0, tfull = 1;

    for (int j0 = 0; j0 < CHUNK; j0 += 32) {
        int pc = (int)cls[base + j0 + lane];
        int tc = (int)tgt[base + j0 + lane];
        int code = pc | (tc << 8);
        #pragma unroll
        for (int j = 0; j < 32; ++j) {
            int cj = __shfl(code, j, 32);    // broadcast position j0+j to all lanes
            int m  = ((cj & 255) == lane);
            prun   = m ? (prun + 1) : 0;
            pbest  = max(pbest, prun);
            ppref += (pfull & m);
            pfull &= m;
            int mt = ((cj >> 8) == lane);
            trun   = mt ? (trun + 1) : 0;
            tbest  = max(tbest, trun);
            tpref += (tfull & mt);
            tfull &= mt;
        }
    }
    tup[0][wave][lane][0] = ppref;  tup[0][wave][lane][1] = prun;
    tup[0][wave][lane][2] = pbest;  tup[0][wave][lane][3] = pfull;
    tup[1][wave][lane][0] = tpref;  tup[1][wave][lane][1] = trun;
    tup[1][wave][lane][2] = tbest;  tup[1][wave][lane][3] = tfull;
    __syncthreads();

    if (threadIdx.x < 32) {
        int finals[2];
        #pragma unroll
        for (int s = 0; s < 2; ++s) {
            int pref = tup[s][0][lane][0];
            int suff = tup[s][0][lane][1];
            int best = tup[s][0][lane][2];
            int full = tup[s][0][lane][3];
            for (int k = 1; k < 32; ++k) {
                int bpref = tup[s][k][lane][0];
                int bsuff = tup[s][k][lane][1];
                int bbest = tup[s][k][lane][2];
                int bfull = tup[s][k][lane][3];
                int cross = suff + bpref;
                best = max(max(best, bbest), cross);
                pref = full  ? (pref + bpref) : pref;
                suff = bfull ? (bsuff + suff) : bsuff;
                full &= bfull;
            }
            finals[s] = best;
        }
        float d  = (float)(finals[0] - finals[1]);
        float sq = d * d;
        #pragma unroll
        for (int off = 16; off > 0; off >>= 1)
            sq += __shfl_down(sq, off, 32);   // fixed-order tree: deterministic
        if (lane == 0)
            rowsum[row] = sq;
    }
}

// ---------------- Kernel 3: fold 128 row sums in fixed order --------
__global__ void fcl_finalize_kernel(const float* __restrict__ rowsum,
                                    float* __restrict__ out)
{
    const int lane = threadIdx.x;                 // one wave of 32
    float s = rowsum[lane]
            + rowsum[lane + 32]
            + rowsum[lane + 64]
            + rowsum[lane + 96];                  // fixed order
    #pragma unroll
    for (int off = 16; off > 0; off >>= 1)
        s += __shfl_down(s, off, 32);             // fixed-order tree
    if (lane == 0)
        out[0] = s * (1.0f / (float)(NB * NC));   // mean over [128, 32]
}

// ---------------- Launch ----------------
extern "C" void kernel_launch(void* const* d_in, const int* in_sizes, int n_in,
                              void* d_out, int out_size, void* d_ws, size_t ws_size,
                              hipStream_t stream) {
    const float*     pred = (const float*)d_in[0];
    const long long* tgt  = (const long long*)d_in[1];   // int64 targets
    float*           out  = (float*)d_out;

    unsigned char* cls    = (unsigned char*)d_ws;                  // 1 MB of u8 classes
    float*         rowsum = (float*)((char*)d_ws + (size_t)NB * NW); // 128 floats after cls

    fcl_argmax_kernel<<<K1_BLOCKS, K1_THREADS, 0, stream>>>(pred, cls);
    fcl_runlen_kernel<<<NB, K2_THREADS, 0, stream>>>(cls, tgt, rowsum);
    fcl_finalize_kernel<<<1, 32, 0, stream>>>(rowsum, out);
}